// GridProjectPooling_4363686772890
// MI455X (gfx1250) — compile-verified
//
#include <hip/hip_runtime.h>

typedef __attribute__((ext_vector_type(2))) float v2f;
typedef __attribute__((ext_vector_type(4))) float v4f;
typedef __attribute__((ext_vector_type(8))) float v8f;

#define GB 4
#define GC 256
#define GH 512
#define GW 512
#define NSEG 32
#define NLINES 31

// ---------------- Kernel A: segmentation tables (1 block, trivial) ----------------
__global__ void gpp_setup(const float* __restrict__ hl, const float* __restrict__ vl,
                          int* __restrict__ rowid, int* __restrict__ colid,
                          unsigned int* __restrict__ cid8, float* __restrict__ inv_area)
{
    __shared__ float sh[NLINES], sv[NLINES];
    __shared__ int   hb[NLINES], vb[NLINES];
    __shared__ float rcnt[NSEG], ccnt[NSEG];
    __shared__ unsigned char cbytes[GW];

    const int t = threadIdx.x;
    if (t < NLINES) { sh[t] = hl[t]; sv[t] = vl[t]; }
    __syncthreads();
    if (t == 0) {  // insertion sort (31 elems) — reference sorts the lines
        for (int i = 1; i < NLINES; ++i) { float k = sh[i]; int j = i - 1;
            while (j >= 0 && sh[j] > k) { sh[j+1] = sh[j]; --j; } sh[j+1] = k; }
        for (int i = 1; i < NLINES; ++i) { float k = sv[i]; int j = i - 1;
            while (j >= 0 && sv[j] > k) { sv[j+1] = sv[j]; --j; } sv[j+1] = k; }
    }
    __syncthreads();
    if (t < NLINES) {
        hb[t] = (int)rintf((float)GH * sh[t]);   // rne, matches jnp.round
        vb[t] = (int)rintf((float)GW * sv[t]);
    }
    __syncthreads();
    if (t < GH) {  // searchsorted(bounds, i, 'right') == count(bounds <= i)
        int r = 0, c = 0;
        for (int k = 0; k < NLINES; ++k) { r += (hb[k] <= t); c += (vb[k] <= t); }
        rowid[t] = r; colid[t] = c; cbytes[t] = (unsigned char)c;
    }
    if (t < NSEG) {
        int lo = t ? hb[t-1] : 0; int hi = (t == NSEG-1) ? GH : hb[t];
        rcnt[t] = fmaxf((float)(hi - lo), 1.0f);
        lo = t ? vb[t-1] : 0; hi = (t == NSEG-1) ? GW : vb[t];
        ccnt[t] = fmaxf((float)(hi - lo), 1.0f);
    }
    __syncthreads();
    if (t < GW / 4) {
        cid8[t] = (unsigned int)cbytes[4*t] |
                  ((unsigned int)cbytes[4*t+1] << 8) |
                  ((unsigned int)cbytes[4*t+2] << 16) |
                  ((unsigned int)cbytes[4*t+3] << 24);
    }
    for (int i = t; i < NSEG * NSEG; i += blockDim.x)
        inv_area[i] = 1.0f / (rcnt[i >> 5] * ccnt[i & 31]);
}

// ---------------- Kernel B: WMMA segmented reduction, 1 block per (b,c) plane ----------------
// matrix[m,n] = sum_{cell} x / (rcount[m]*ccount[n]).
// Each wave owns 4 w-tiles (fused in one K-loop so the one-hot A operand and the
// row-id LDS reads are built ONCE per 4-row chunk and shared by 8 WMMA chains):
//   S[rowseg, w] += OneHot(16x4) x Xtile(4x16)  via v_wmma_f32_16x16x4_f32
// then S collapses over w by col-segment with LDS float atomics.
__global__ void __launch_bounds__(256)
gpp_reduce(const float* __restrict__ x, const int* __restrict__ rowid,
           const int* __restrict__ colid, const float* __restrict__ inv_area,
           float* __restrict__ out_mat)
{
    __shared__ float s_cell[NSEG * NSEG];
    __shared__ int   s_rid[GH];
    __shared__ int   s_cid[GW];

    const int plane = blockIdx.x;                       // b*C + c
    const float* __restrict__ xp = x + (size_t)plane * (GH * GW);
    const int tid = threadIdx.x;

    for (int i = tid; i < NSEG * NSEG; i += 256) s_cell[i] = 0.0f;
    for (int i = tid; i < GH; i += 256) s_rid[i] = rowid[i];
    for (int i = tid; i < GW; i += 256) s_cid[i] = colid[i];
    __syncthreads();

    const int wave  = tid >> 5;        // wave32
    const int lane  = tid & 31;
    const int lhalf = lane >> 4;       // 0: covers K=0,1 ; 1: covers K=2,3
    const int l15   = lane & 15;
    const int wbase = wave * 16;       // wave's tile j sits at w0 = wbase + j*128

    v8f acc_lo[4] = {{0,0,0,0,0,0,0,0},{0,0,0,0,0,0,0,0},
                     {0,0,0,0,0,0,0,0},{0,0,0,0,0,0,0,0}};  // segments 0..15
    v8f acc_hi[4] = {{0,0,0,0,0,0,0,0},{0,0,0,0,0,0,0,0},
                     {0,0,0,0,0,0,0,0},{0,0,0,0,0,0,0,0}};  // segments 16..31

    // B operand base: lane half picks rows h0+{0,2}; +GW is the next row.
    const float* bptr = xp + (size_t)(lhalf * 2) * GW + (wbase + l15);

    for (int h0 = 0; h0 < GH; h0 += 4) {
        const int ra = s_rid[h0 + lhalf * 2];
        const int rb = s_rid[h0 + lhalf * 2 + 1];
        v2f a_lo, a_hi;                                 // A: one-hot row-seg (16x4)
        a_lo.x = (ra == l15)      ? 1.0f : 0.0f;
        a_lo.y = (rb == l15)      ? 1.0f : 0.0f;
        a_hi.x = (ra == l15 + 16) ? 1.0f : 0.0f;
        a_hi.y = (rb == l15 + 16) ? 1.0f : 0.0f;
#pragma unroll
        for (int j = 0; j < 4; ++j) {                   // 4 w-tiles share A
            v2f bm;                                     // B: 4x16 tile of x
            bm.x = bptr[j * 128];                       // K = 2*lhalf
            bm.y = bptr[GW + j * 128];                  // K = 2*lhalf + 1
            acc_lo[j] = __builtin_amdgcn_wmma_f32_16x16x4_f32(
                false, a_lo, false, bm, (short)0, acc_lo[j], false, false);
            acc_hi[j] = __builtin_amdgcn_wmma_f32_16x16x4_f32(
                false, a_hi, false, bm, (short)0, acc_hi[j], false, false);
        }
        bptr += 4 * GW;
    }

    // D layout: vgpr r, lane<16 -> (M=r, N=lane), lane>=16 -> (M=r+8, N=lane-16)
    const int segbase = lhalf * 8;
#pragma unroll
    for (int j = 0; j < 4; ++j) {
        const int n = s_cid[wbase + j * 128 + l15];
#pragma unroll
        for (int r = 0; r < 8; ++r) {
            atomicAdd(&s_cell[(segbase + r) * NSEG + n],      acc_lo[j][r]);
            atomicAdd(&s_cell[(16 + segbase + r) * NSEG + n], acc_hi[j][r]);
        }
    }
    __syncthreads();
    for (int i = tid; i < NSEG * NSEG; i += 256)
        out_mat[(size_t)plane * (NSEG * NSEG) + i] = s_cell[i] * inv_area[i];
}

// ---------------- Kernel C: broadcast cell means back to pixels (write-streaming) ----------------
__global__ void __launch_bounds__(256)
gpp_bcast(const float* __restrict__ mat, const int* __restrict__ rowid,
          const unsigned int* __restrict__ cid8, float* __restrict__ out)
{
    __shared__ float srow[2][NSEG];
    const int blk   = blockIdx.x;
    const int plane = blk >> 8;            // 256 row-pairs per plane
    const int hp    = (blk & 255) << 1;
    const int t     = threadIdx.x;

    if (t < 2 * NSEG) {                    // stage the two needed matrix rows
        const int rr = t >> 5;
        const int m  = rowid[hp + rr];
        srow[rr][t & 31] = mat[(size_t)plane * (NSEG * NSEG) + m * NSEG + (t & 31)];
    }
    __syncthreads();

    const int sub = t >> 7;                // which of the 2 rows
    const int lw  = t & 127;               // float4 index within the row
    const unsigned int c4 = cid8[lw];      // 4 packed col ids
    v4f v;
    v.x = srow[sub][c4 & 255];
    v.y = srow[sub][(c4 >> 8) & 255];
    v.z = srow[sub][(c4 >> 16) & 255];
    v.w = srow[sub][(c4 >> 24)];
    const size_t off = (((size_t)plane * GH) + (hp + sub)) * GW + (size_t)lw * 4;
    __builtin_nontemporal_store(v, (v4f*)(out + off));
}

extern "C" void kernel_launch(void* const* d_in, const int* in_sizes, int n_in,
                              void* d_out, int out_size, void* d_ws, size_t ws_size,
                              hipStream_t stream) {
    const float* x  = (const float*)d_in[0];   // (4,256,512,512) f32
    const float* hl = (const float*)d_in[1];   // (31,) f32
    const float* vl = (const float*)d_in[2];   // (31,) f32
    float* out     = (float*)d_out;                               // output region
    float* out_mat = out + (size_t)GB * GC * GH * GW;             // matrix region

    int* rowid          = (int*)d_ws;                 // 512 i32
    int* colid          = rowid + GH;                 // 512 i32
    unsigned int* cid8  = (unsigned int*)(colid + GW);// 128 u32 (packed bytes)
    float* inv_area     = (float*)(cid8 + GW / 4);    // 1024 f32

    gpp_setup <<<1, 512, 0, stream>>>(hl, vl, rowid, colid, cid8, inv_area);
    gpp_reduce<<<GB * GC, 256, 0, stream>>>(x, rowid, colid, inv_area, out_mat);
    gpp_bcast <<<GB * GC * (GH / 2), 256, 0, stream>>>(out_mat, rowid, cid8, out);
}